// BarcodeSLayerEncoder_20486994002574
// MI455X (gfx1250) — compile-verified
//
#include <hip/hip_runtime.h>
#include <math.h>

// Problem constants (match reference)
#define B_    256
#define P_    2048
#define E_    16
#define HID_  128
#define OUT_  128
#define BN_EPS 1e-5f

typedef float v2f __attribute__((ext_vector_type(2)));
typedef float v8f __attribute__((ext_vector_type(8)));

__device__ __forceinline__ float softplus_f(float x) {
    // numerically stable softplus; matches jax.nn.softplus semantics
    return (x > 20.0f) ? x : log1pf(__expf(x));
}

// ---------------------------------------------------------------------------
// SLayer: per batch b, sum over valid points of exp(-sum_d sharp*(pt-center)^2)
// grid = B blocks, 256 threads (8 wave32s). Centers/sharpness live in VGPRs.
// ---------------------------------------------------------------------------
__global__ __launch_bounds__(256) void slayer_kernel(
    const float* __restrict__ pts,       // [B,P,2]
    const int*   __restrict__ counts,    // [B]
    const float* __restrict__ centers,   // [E,2]
    const float* __restrict__ log_sharp, // [E,2]
    float* __restrict__ xcol)            // write xcol[b*32 + e]
{
    const int b   = blockIdx.x;
    const int tid = threadIdx.x;

    float cx[E_], cy[E_], sx[E_], sy[E_], acc[E_];
#pragma unroll
    for (int e = 0; e < E_; ++e) {
        cx[e] = centers[2*e + 0];
        cy[e] = centers[2*e + 1];
        sx[e] = softplus_f(log_sharp[2*e + 0]) + 1e-6f;
        sy[e] = softplus_f(log_sharp[2*e + 1]) + 1e-6f;
        acc[e] = 0.0f;
    }

    const int cnt = counts[b];
    const float* __restrict__ pb = pts + (size_t)b * P_ * 2;
    for (int p = tid; p < cnt; p += 256) {
        const float px = pb[2*p + 0];
        const float py = pb[2*p + 1];
#pragma unroll
        for (int e = 0; e < E_; ++e) {
            const float dx = px - cx[e];
            const float dy = py - cy[e];
            acc[e] += __expf(-(sx[e]*dx*dx + sy[e]*dy*dy));
        }
    }

    // wave32 reduction
#pragma unroll
    for (int e = 0; e < E_; ++e) {
#pragma unroll
        for (int off = 16; off > 0; off >>= 1)
            acc[e] += __shfl_xor(acc[e], off, 32);
    }

    __shared__ float red[E_][8];
    const int wave = tid >> 5;
    const int lane = tid & 31;
    if (lane == 0) {
#pragma unroll
        for (int e = 0; e < E_; ++e) red[e][wave] = acc[e];
    }
    __syncthreads();
    if (tid < E_) {
        float s = 0.0f;
#pragma unroll
        for (int w = 0; w < 8; ++w) s += red[tid][w];
        xcol[b * 32 + tid] = s;
    }
}

// ---------------------------------------------------------------------------
// fp32 GEMM via V_WMMA_F32_16X16X4_F32.
// M=256, N=128 fixed; K templated (32 or 128). 16 blocks x 256 threads =
// 128 wave32s, one 16x16 D-tile per wave. BN_RELU is a compile-time flag:
// when set, the A operand gets a precomputed affine BN (fma) + ReLU on load.
// Uniform control flow everywhere: EXEC stays all-1s as WMMA requires.
//
// Fragment layouts per CDNA5 ISA 7.12.2:
//   A 16x4 : lanes 0-15 -> K=kk+{0,1} in v0,v1 ; lanes 16-31 -> K=kk+{2,3}
//   B 4x16 : same K split, N striped across 16 lanes within each VGPR
//   C/D    : v[r] = row m0+r (lanes 0-15) / m0+r+8 (lanes 16-31), col n0+lane
// ---------------------------------------------------------------------------
template<int K, bool BN_RELU>
__global__ __launch_bounds__(256) void gemm_wmma_kernel(
    const float* __restrict__ A,     // [256, K]
    const float* __restrict__ Bm,    // [K, 128]
    float* __restrict__ Dm,          // [256, 128]
    const float* __restrict__ scale, // [K]  (rstd*gamma)      if BN_RELU
    const float* __restrict__ shift) // [K]  (beta-mean*scale) if BN_RELU
{
    const int w    = blockIdx.x * 8 + (threadIdx.x >> 5); // 0..127
    const int m0   = (w >> 3) << 4;                       // 16 M-tiles
    const int n0   = (w & 7) << 4;                        // 8 N-tiles
    const int lane = threadIdx.x & 31;
    const int half = lane >> 4;                           // 0 or 1
    const int l    = lane & 15;

    v8f c = {};
#pragma unroll
    for (int kk = 0; kk < K; kk += 4) {
        const int ka = kk + 2 * half;

        float a0 = A[(m0 + l) * K + ka];
        float a1 = A[(m0 + l) * K + ka + 1];
        if (BN_RELU) {
            a0 = fmaxf(fmaf(a0, scale[ka],     shift[ka]),     0.0f);
            a1 = fmaxf(fmaf(a1, scale[ka + 1], shift[ka + 1]), 0.0f);
        }
        v2f a; a.x = a0; a.y = a1;

        v2f bfrag;
        bfrag.x = Bm[(ka)     * 128 + n0 + l];
        bfrag.y = Bm[(ka + 1) * 128 + n0 + l];

        // 8 args: (neg_a, A, neg_b, B, c_mod, C, reuse_a, reuse_b)
        c = __builtin_amdgcn_wmma_f32_16x16x4_f32(
                false, a, false, bfrag, (short)0, c, false, false);
    }

#pragma unroll
    for (int r = 0; r < 8; ++r) {
        Dm[(m0 + r + 8 * half) * 128 + n0 + l] = c[r];
    }
}

// ---------------------------------------------------------------------------
// Column batch-norm stats over 256 rows x 128 cols (biased variance), fused
// with the affine params: scale[c] = rstd*gamma, shift[c] = beta - mean*scale.
// ---------------------------------------------------------------------------
__global__ __launch_bounds__(128) void colstats_kernel(
    const float* __restrict__ X,     // [256,128]
    const float* __restrict__ gamma, // [128]
    const float* __restrict__ beta,  // [128]
    float* __restrict__ scale, float* __restrict__ shift)
{
    const int c = threadIdx.x;
    float s = 0.0f, s2 = 0.0f;
    for (int r = 0; r < 256; ++r) {
        const float v = X[r * 128 + c];
        s  += v;
        s2 += v * v;
    }
    const float m   = s * (1.0f / 256.0f);
    const float var = s2 * (1.0f / 256.0f) - m * m;
    const float sc  = rsqrtf(var + BN_EPS) * gamma[c];
    scale[c] = sc;
    shift[c] = fmaf(-m, sc, beta[c]);
}

// ---------------------------------------------------------------------------
// BN2 (affine form) + per-row L2 normalize. grid = 256 rows, 128 threads.
// ---------------------------------------------------------------------------
__global__ __launch_bounds__(128) void finalize_kernel(
    const float* __restrict__ Y,     // [256,128]
    const float* __restrict__ scale,
    const float* __restrict__ shift,
    float* __restrict__ out)         // [256,128]
{
    const int b = blockIdx.x;
    const int c = threadIdx.x;

    const float v = fmaf(Y[b * 128 + c], scale[c], shift[c]);

    float ss = v * v;
#pragma unroll
    for (int off = 16; off > 0; off >>= 1)
        ss += __shfl_xor(ss, off, 32);

    __shared__ float wred[4];
    __shared__ float inv_norm;
    const int wave = c >> 5;
    const int lane = c & 31;
    if (lane == 0) wred[wave] = ss;
    __syncthreads();
    if (c == 0) {
        const float total = wred[0] + wred[1] + wred[2] + wred[3];
        inv_norm = 1.0f / fmaxf(sqrtf(total), 1e-12f);
    }
    __syncthreads();
    out[b * 128 + c] = v * inv_norm;
}

// ---------------------------------------------------------------------------
extern "C" void kernel_launch(void* const* d_in, const int* in_sizes, int n_in,
                              void* d_out, int out_size, void* d_ws, size_t ws_size,
                              hipStream_t stream) {
    const float* h0   = (const float*)d_in[0];
    const int*   cnt0 = (const int*)  d_in[1];
    const float* h1   = (const float*)d_in[2];
    const int*   cnt1 = (const int*)  d_in[3];
    const float* c0   = (const float*)d_in[4];
    const float* ls0  = (const float*)d_in[5];
    const float* c1   = (const float*)d_in[6];
    const float* ls1  = (const float*)d_in[7];
    const float* W1   = (const float*)d_in[8];
    const float* g1   = (const float*)d_in[9];
    const float* be1  = (const float*)d_in[10];
    const float* W2   = (const float*)d_in[11];
    const float* g2   = (const float*)d_in[12];
    const float* be2  = (const float*)d_in[13];
    float* out = (float*)d_out;

    // Workspace layout (fp32 elements)
    float* x      = (float*)d_ws;          // [256,32]
    float* h      = x + 256 * 32;          // [256,128]
    float* y      = h + 256 * 128;         // [256,128]
    float* scale1 = y + 256 * 128;         // [128]
    float* shift1 = scale1 + 128;          // [128]
    float* scale2 = shift1 + 128;          // [128]
    float* shift2 = scale2 + 128;          // [128]

    // SLayer features (h0 -> cols 0..15, h1 -> cols 16..31)
    slayer_kernel<<<B_, 256, 0, stream>>>(h0, cnt0, c0, ls0, x);
    slayer_kernel<<<B_, 256, 0, stream>>>(h1, cnt1, c1, ls1, x + 16);

    // GEMM1: x[256,32] @ W1[32,128] -> h_raw
    gemm_wmma_kernel<32, false><<<16, 256, 0, stream>>>(
        x, W1, h, nullptr, nullptr);

    // BN1 stats -> fused affine params
    colstats_kernel<<<1, 128, 0, stream>>>(h, g1, be1, scale1, shift1);

    // GEMM2: relu(bn1(h_raw)) @ W2[128,128] -> y_raw (BN+ReLU fused on A load)
    gemm_wmma_kernel<128, true><<<16, 256, 0, stream>>>(
        h, W2, y, scale1, shift1);

    // BN2 stats -> fused affine params
    colstats_kernel<<<1, 128, 0, stream>>>(y, g2, be2, scale2, shift2);

    // BN2 + L2 row-normalize -> out
    finalize_kernel<<<B_, 128, 0, stream>>>(y, scale2, shift2, out);
}